// EdgeClassifier_25220047962748
// MI455X (gfx1250) — compile-verified
//
#include <hip/hip_runtime.h>

typedef __attribute__((ext_vector_type(2))) float v2f;
typedef __attribute__((ext_vector_type(8))) float v8f;

static __device__ __forceinline__ v8f wmma_f32_4(v2f a, v2f b, v8f c) {
  // D = A(16x4 f32) x B(4x16 f32) + C(16x16 f32), wave32
  return __builtin_amdgcn_wmma_f32_16x16x4_f32(false, a, false, b, (short)0, c,
                                               false, false);
}

// ---------------- CSR build: histogram / scan / scatter ----------------
__global__ void zero_counts(int* cnt, int n) {
  int i = blockIdx.x * blockDim.x + threadIdx.x;
  if (i < n) cnt[i] = 0;
}

__global__ void count_dst(int* cnt, const int* __restrict__ dst, int e) {
  int i = blockIdx.x * blockDim.x + threadIdx.x;
  if (i < e) atomicAdd(&cnt[dst[i]], 1);
}

__global__ void dinv_from_counts(float* dinv, const int* __restrict__ cnt, int n) {
  int i = blockIdx.x * blockDim.x + threadIdx.x;
  if (i < n) dinv[i] = rsqrtf((float)cnt[i] + 1.0f);  // +1 self loop
}

// single-block exclusive scan; overwrites cnt_cursor with row starts,
// writes rowptr[0..n] (rowptr[n] = E). cnt_cursor intentionally NOT restrict
// (read counts, write cursor through the same pointer, per-element RAW only).
__global__ __launch_bounds__(1024) void scan_counts(int* cnt_cursor,
                                                    int* __restrict__ rowptr,
                                                    int n) {
  __shared__ int sdata[1024];
  __shared__ int scarry;
  int tid = threadIdx.x;
  if (tid == 0) scarry = 0;
  __syncthreads();
  for (int base = 0; base < n; base += 1024) {
    int i = base + tid;
    int v = (i < n) ? cnt_cursor[i] : 0;
    sdata[tid] = v;
    __syncthreads();
    for (int off = 1; off < 1024; off <<= 1) {
      int t = (tid >= off) ? sdata[tid - off] : 0;
      __syncthreads();
      sdata[tid] += t;
      __syncthreads();
    }
    int incl = sdata[tid];
    int excl = incl - v;
    if (i < n) {
      int start = scarry + excl;
      rowptr[i] = start;
      cnt_cursor[i] = start;  // cursor init for scatter
    }
    __syncthreads();
    if (tid == 1023) scarry += sdata[1023];
    __syncthreads();
  }
  if (tid == 0) rowptr[n] = scarry;
}

__global__ void scatter_edges(int* cursor, int* __restrict__ eidx,
                              const int* __restrict__ dst, int e) {
  int i = blockIdx.x * blockDim.x + threadIdx.x;
  if (i < e) {
    int pos = atomicAdd(&cursor[dst[i]], 1);
    eidx[pos] = i;
  }
}

// ---------------- GCN GEMM: one wave per 16-row tile ----------------
// traw[row][col] = (relu?(A)) @ W   (raw product; bias/norm applied in gather)
template <int K, bool RELU_A>
__global__ __launch_bounds__(128) void gemm_gcn(
    const float* __restrict__ A, const float* __restrict__ W,
    float* __restrict__ traw, int ntiles) {
  int wave = blockIdx.x * (blockDim.x >> 5) + (threadIdx.x >> 5);
  if (wave >= ntiles) return;  // wave-uniform: EXEC all-ones at WMMA
  int lane = threadIdx.x & 31;
  int m = lane & 15;
  int half = lane >> 4;
  int row0 = wave * 16;

  v8f acc[4] = {};  // 4 column tiles of 16 (H=64)

  const float* arow = A + (size_t)(row0 + m) * K + 2 * half;
#pragma unroll
  for (int kb = 0; kb < K; kb += 4) {
    float a0 = arow[kb];
    float a1 = arow[kb + 1];
    if (RELU_A) { a0 = fmaxf(a0, 0.0f); a1 = fmaxf(a1, 0.0f); }
    v2f a; a.x = a0; a.y = a1;
    int k0 = kb + 2 * half;
#pragma unroll
    for (int nt = 0; nt < 4; ++nt) {
      int col = nt * 16 + m;
      v2f b; b.x = W[k0 * 64 + col]; b.y = W[(k0 + 1) * 64 + col];
      acc[nt] = wmma_f32_4(a, b, acc[nt]);
    }
  }

#pragma unroll
  for (int r = 0; r < 8; ++r) {
    int row = row0 + r + 8 * half;
#pragma unroll
    for (int nt = 0; nt < 4; ++nt)
      traw[(size_t)row * 64 + nt * 16 + m] = acc[nt][r];
  }
}

// ---------------- aggregation: one wave per node, CSR gather ----------------
// out[v][c] = bias[c] + dinv[v]^2 * traw[v][c]
//           + sum_{e: dst==v} dinv[src]*dinv[v] * traw[src][c]
__global__ __launch_bounds__(128) void agg_gather(
    const float* __restrict__ traw, float* __restrict__ outp,
    const int* __restrict__ rowptr, const int* __restrict__ eidx,
    const int* __restrict__ src, const float* __restrict__ dinv,
    const float* __restrict__ bias, int n) {
  int v = blockIdx.x * (blockDim.x >> 5) + (threadIdx.x >> 5);
  if (v >= n) return;
  int lane = threadIdx.x & 31;
  float dv = dinv[v];
  const float* trow = traw + (size_t)v * 64;
  float a0 = bias[lane]      + dv * dv * trow[lane];
  float a1 = bias[lane + 32] + dv * dv * trow[lane + 32];
  int jb = rowptr[v], je = rowptr[v + 1];
  for (int j = jb; j < je; ++j) {
    int e = eidx[j];
    int s = src[e];
    float w = dinv[s] * dv;
    const float* srow = traw + (size_t)s * 64;
    a0 += w * srow[lane];
    a1 += w * srow[lane + 32];
  }
  outp[(size_t)v * 64 + lane]      = a0;
  outp[(size_t)v * 64 + lane + 32] = a1;
}

// ---------------- edge MLP: one wave per 16 edges ----------------
#define EMLP_WAVES 4
__global__ __launch_bounds__(32 * EMLP_WAVES) void edge_mlp(
    const float* __restrict__ h2, const float* __restrict__ eattr,
    const int* __restrict__ src, const int* __restrict__ dst,
    const float* __restrict__ Wc1, const float* __restrict__ bc1,
    const float* __restrict__ Wc2, const float* __restrict__ bc2,
    float* __restrict__ out, int ntiles) {
  __shared__ float sg[EMLP_WAVES][16][68];  // padded: avoid bank conflicts
  int wslot = threadIdx.x >> 5;
  int wave = blockIdx.x * EMLP_WAVES + wslot;
  if (wave >= ntiles) return;  // uniform per wave
  int lane = threadIdx.x & 31;
  int m = lane & 15;
  int half = lane >> 4;
  int e = wave * 16 + m;
  int s = src[e], d = dst[e];

  // acc starts at bc1 (column bias, same for every row)
  v8f acc[4];
#pragma unroll
  for (int nt = 0; nt < 4; ++nt) {
    float bv = bc1[nt * 16 + m];
#pragma unroll
    for (int r = 0; r < 8; ++r) acc[nt][r] = bv;
  }

  const float* pS = h2 + (size_t)s * 64 + 2 * half;
  const float* pD = h2 + (size_t)d * 64 + 2 * half;
  const float* pE = eattr + (size_t)e * 16 + 2 * half;

  // K segment 1: h2[src], k = 0..63
#pragma unroll 4
  for (int kb = 0; kb < 64; kb += 4) {
    v2f a; a.x = pS[kb]; a.y = pS[kb + 1];
    int k0 = kb + 2 * half;
#pragma unroll
    for (int nt = 0; nt < 4; ++nt) {
      v2f b; b.x = Wc1[k0 * 64 + nt * 16 + m]; b.y = Wc1[(k0 + 1) * 64 + nt * 16 + m];
      acc[nt] = wmma_f32_4(a, b, acc[nt]);
    }
  }
  // K segment 2: h2[dst], k = 64..127
#pragma unroll 4
  for (int kb = 0; kb < 64; kb += 4) {
    v2f a; a.x = pD[kb]; a.y = pD[kb + 1];
    int k0 = 64 + kb + 2 * half;
#pragma unroll
    for (int nt = 0; nt < 4; ++nt) {
      v2f b; b.x = Wc1[k0 * 64 + nt * 16 + m]; b.y = Wc1[(k0 + 1) * 64 + nt * 16 + m];
      acc[nt] = wmma_f32_4(a, b, acc[nt]);
    }
  }
  // K segment 3: edge_attr, k = 128..143
#pragma unroll
  for (int kb = 0; kb < 16; kb += 4) {
    v2f a; a.x = pE[kb]; a.y = pE[kb + 1];
    int k0 = 128 + kb + 2 * half;
#pragma unroll
    for (int nt = 0; nt < 4; ++nt) {
      v2f b; b.x = Wc1[k0 * 64 + nt * 16 + m]; b.y = Wc1[(k0 + 1) * 64 + nt * 16 + m];
      acc[nt] = wmma_f32_4(a, b, acc[nt]);
    }
  }

  // ReLU -> LDS tile (row-major 16x64, stride 68)
  float(*sgw)[68] = sg[wslot];
#pragma unroll
  for (int r = 0; r < 8; ++r) {
    int row = r + 8 * half;
#pragma unroll
    for (int nt = 0; nt < 4; ++nt)
      sgw[row][nt * 16 + m] = fmaxf(acc[nt][r], 0.0f);
  }
  __builtin_amdgcn_wave_barrier();  // in-wave LDS write->read ordering point

  // second layer: [16,64] @ [64,5] + bc2, split K across wave halves
  float o0 = 0.f, o1 = 0.f, o2 = 0.f, o3 = 0.f, o4 = 0.f;
  const float* grow = sgw[m];
  int kbeg = 32 * half;
#pragma unroll 8
  for (int k = kbeg; k < kbeg + 32; ++k) {
    float g = grow[k];
    const float* wr = Wc2 + k * 5;
    o0 += g * wr[0];
    o1 += g * wr[1];
    o2 += g * wr[2];
    o3 += g * wr[3];
    o4 += g * wr[4];
  }
  o0 += __shfl_down(o0, 16);
  o1 += __shfl_down(o1, 16);
  o2 += __shfl_down(o2, 16);
  o3 += __shfl_down(o3, 16);
  o4 += __shfl_down(o4, 16);
  if (half == 0) {
    float* op = out + (size_t)e * 5;
    op[0] = o0 + bc2[0];
    op[1] = o1 + bc2[1];
    op[2] = o2 + bc2[2];
    op[3] = o3 + bc2[3];
    op[4] = o4 + bc2[4];
  }
}

// ---------------- host launch ----------------
extern "C" void kernel_launch(void* const* d_in, const int* in_sizes, int n_in,
                              void* d_out, int out_size, void* d_ws, size_t ws_size,
                              hipStream_t stream) {
  const float* x     = (const float*)d_in[0];   // [N,32]
  const int*   ei    = (const int*)d_in[1];     // [2,E]
  const float* eattr = (const float*)d_in[2];   // [E,16]
  const float* W1    = (const float*)d_in[3];   // [32,64]
  const float* b1    = (const float*)d_in[4];
  const float* W2    = (const float*)d_in[5];   // [64,64]
  const float* b2    = (const float*)d_in[6];
  const float* Wc1   = (const float*)d_in[7];   // [144,64]
  const float* bc1   = (const float*)d_in[8];
  const float* Wc2   = (const float*)d_in[9];   // [64,5]
  const float* bc2   = (const float*)d_in[10];
  float* out = (float*)d_out;

  int N = in_sizes[0] / 32;
  int E = in_sizes[1] / 2;
  const int* src = ei;
  const int* dst = ei + E;

  float* wf     = (float*)d_ws;
  float* dinv   = wf;                       // [N]
  float* tA     = wf + N;                   // [N,64] raw GEMM products
  float* tB     = tA + (size_t)N * 64;      // [N,64] aggregated activations
  int*   rowptr = (int*)(tB + (size_t)N * 64);  // [N+1]
  int*   cursor = rowptr + (N + 1);         // [N] counts -> cursors
  int*   eidx   = cursor + N;               // [E]

  int gN = (N + 255) / 256;
  int gE = (E + 255) / 256;

  // CSR by dst (once; reused for both GCN layers)
  zero_counts<<<gN, 256, 0, stream>>>(cursor, N);
  count_dst<<<gE, 256, 0, stream>>>(cursor, dst, E);
  dinv_from_counts<<<gN, 256, 0, stream>>>(dinv, cursor, N);
  scan_counts<<<1, 1024, 0, stream>>>(cursor, rowptr, N);
  scatter_edges<<<gE, 256, 0, stream>>>(cursor, eidx, dst, E);

  int ntiles = (N + 15) / 16;   // N = 100000 -> 6250 exact
  int nwBlocks = (N + 3) / 4;   // agg_gather: 4 waves/block

  // layer 1
  gemm_gcn<32, false><<<(ntiles + 3) / 4, 128, 0, stream>>>(x, W1, tA, ntiles);
  agg_gather<<<nwBlocks, 128, 0, stream>>>(tA, tB, rowptr, eidx, src, dinv, b1, N);

  // layer 2 (ReLU fused into A-load)
  gemm_gcn<64, true><<<(ntiles + 3) / 4, 128, 0, stream>>>(tB, W2, tA, ntiles);
  agg_gather<<<nwBlocks, 128, 0, stream>>>(tA, tB, rowptr, eidx, src, dinv, b2, N);

  // edge MLP on h2 = tB
  int etiles = (E + 15) / 16;   // E = 1.6M -> 100000 exact
  edge_mlp<<<(etiles + EMLP_WAVES - 1) / EMLP_WAVES, 32 * EMLP_WAVES, 0, stream>>>(
      tB, eattr, src, dst, Wc1, bc1, Wc2, bc2, out, etiles);
}